// continuousB_86895778333460
// MI455X (gfx1250) — compile-verified
//
#include <hip/hip_runtime.h>

typedef __attribute__((ext_vector_type(2))) float v2f;
typedef __attribute__((ext_vector_type(8))) float v8f;

#define NMAT 4096
#define MELEM (NMAT * NMAT)

// ---------------- zero init (grid-stride, float4) ----------------
__global__ void k_zero_f4(float4* p, int n4) {
  int i = blockIdx.x * blockDim.x + threadIdx.x;
  int stride = gridDim.x * blockDim.x;
  float4 z = {0.f, 0.f, 0.f, 0.f};
  for (; i < n4; i += stride) p[i] = z;
}

// ---------------- scatter edges -> A (stat), W (weights) ----------------
__global__ void k_scatter_edges(const int* __restrict__ edges,
                                const float* __restrict__ wts,
                                const float* __restrict__ stat,
                                float* A, float* W, int E) {
  int e = blockIdx.x * blockDim.x + threadIdx.x;
  if (e >= E) return;
  int r = edges[2 * e], c = edges[2 * e + 1];
  int idx = r * NMAT + c;
  atomicAdd(&W[idx], wts[e]);
  atomicAdd(&A[idx], stat[e]);
}

// ---------------- scatter edges_c -> B (grdt) ----------------
__global__ void k_scatter_edges_c(const int* __restrict__ edges_c,
                                  const float* __restrict__ grdt,
                                  float* B, int EC) {
  int e = blockIdx.x * blockDim.x + threadIdx.x;
  if (e >= EC) return;
  int r = edges_c[2 * e], c = edges_c[2 * e + 1];
  atomicAdd(&B[r * NMAT + c], grdt[e]);
}

// ------- elementwise: V = B*(A-1)+A*W (into B buf), Sm05 = sigmoid(6|V|)-0.5 (into A buf)
__device__ __forceinline__ float sig6m05(float v) {
  // sigmoid(6*|v|) - 0.5  ; arg of exp is <= 0 -> no overflow
  return 1.0f / (1.0f + __expf(-6.0f * fabsf(v))) - 0.5f;
}

__global__ void k_combine(float* A, const float* __restrict__ W, float* Bv, int n4) {
  int i = blockIdx.x * blockDim.x + threadIdx.x;
  if (i >= n4) return;
  float4 a = ((const float4*)A)[i];
  float4 w = ((const float4*)W)[i];
  float4 b = ((const float4*)Bv)[i];
  float4 v, s;
  v.x = b.x * (a.x - 1.0f) + a.x * w.x;
  v.y = b.y * (a.y - 1.0f) + a.y * w.y;
  v.z = b.z * (a.z - 1.0f) + a.z * w.z;
  v.w = b.w * (a.w - 1.0f) + a.w * w.w;
  s.x = sig6m05(v.x);
  s.y = sig6m05(v.y);
  s.z = sig6m05(v.z);
  s.w = sig6m05(v.w);
  ((float4*)Bv)[i] = v;  // V
  ((float4*)A)[i] = s;   // S - 0.5
}

// ------- column sums via WMMA: D = ones(16x4) x B_tile(4x16), accumulate down rows
// wave -> (col stripe = wave&255)*16, row block = (wave>>8)*128
__global__ void k_colsum_wmma(const float* __restrict__ V,
                              const float* __restrict__ S,
                              float* Gacc, float* Ninp) {
  int wave = blockIdx.x * (blockDim.x >> 5) + (threadIdx.x >> 5);
  int lane = threadIdx.x & 31;
  int c0 = (wave & 255) * 16;
  int r0 = (wave >> 8) * 128;
  int col = c0 + (lane & 15);
  int kb = (lane >> 4) * 2;  // lanes 0-15: K=0,1 ; lanes 16-31: K=2,3
  v2f ones = {1.0f, 1.0f};
  v8f accV = {};
  v8f accS = {};
  for (int r = r0; r < r0 + 128; r += 4) {
    int i0 = (r + kb) * NMAT + col;
    v2f bv = {V[i0], V[i0 + NMAT]};
    v2f bs = {S[i0], S[i0 + NMAT]};
    accV = __builtin_amdgcn_wmma_f32_16x16x4_f32(false, ones, false, bv,
                                                 (short)0, accV, false, false);
    accS = __builtin_amdgcn_wmma_f32_16x16x4_f32(false, ones, false, bs,
                                                 (short)0, accS, false, false);
  }
  // every D row identical => element 0 holds the column sum for N = lane (lanes 0-15)
  if (lane < 16) {
    atomicAdd(&Gacc[col], accV[0]);
    atomicAdd(&Ninp[col], accS[0]);
  }
}

// ------- row sums via WMMA: D = A_tile(16x4) x ones(4x16), accumulate across cols
// wave -> (row stripe = wave&255)*16, col block = (wave>>8)*128
__global__ void k_rowsum_wmma(const float* __restrict__ S, float* Noutp) {
  int wave = blockIdx.x * (blockDim.x >> 5) + (threadIdx.x >> 5);
  int lane = threadIdx.x & 31;
  int r0 = (wave & 255) * 16;
  int cb = (wave >> 8) * 128;
  int row = r0 + (lane & 15);
  int coff = (lane >> 4) * 2;  // lanes 0-15: K=0,1 ; lanes 16-31: K=2,3
  v2f ones = {1.0f, 1.0f};
  v8f acc = {};
  for (int c = cb; c < cb + 128; c += 4) {
    const float2 a2 = *(const float2*)(S + row * NMAT + c + coff);
    v2f a = {a2.x, a2.y};
    acc = __builtin_amdgcn_wmma_f32_16x16x4_f32(false, a, false, ones,
                                                (short)0, acc, false, false);
  }
  // D columns identical; lane 0 carries rows r0..r0+7 (M=0..7), lane 16 rows r0+8..r0+15
  if ((lane & 15) == 0) {
    int base = r0 + ((lane >> 4) << 3);
    atomicAdd(&Noutp[base + 0], acc[0]);
    atomicAdd(&Noutp[base + 1], acc[1]);
    atomicAdd(&Noutp[base + 2], acc[2]);
    atomicAdd(&Noutp[base + 3], acc[3]);
    atomicAdd(&Noutp[base + 4], acc[4]);
    atomicAdd(&Noutp[base + 5], acc[5]);
    atomicAdd(&Noutp[base + 6], acc[6]);
    atomicAdd(&Noutp[base + 7], acc[7]);
  }
}

// ------- good[j] = clip(g[j] / (2048 + Ninp[j]), -1, 1)   (N_in >= 2048, never 0)
__global__ void k_good(const float* __restrict__ Gacc, const float* __restrict__ Ninp,
                       float* out_good) {
  int j = blockIdx.x * blockDim.x + threadIdx.x;
  if (j >= NMAT) return;
  float nin = 2048.0f + Ninp[j];
  float g = Gacc[j] / nin;
  out_good[j] = fminf(1.0f, fmaxf(-1.0f, g));
}

// ------- f[i] = sum_j [V!=0] * (1 - |V - good[j]|/2), WMMA row-reduced
__global__ void k_fpass_wmma(const float* __restrict__ V,
                             const float* __restrict__ good, float* Facc) {
  int wave = blockIdx.x * (blockDim.x >> 5) + (threadIdx.x >> 5);
  int lane = threadIdx.x & 31;
  int r0 = (wave & 255) * 16;
  int cb = (wave >> 8) * 128;
  int row = r0 + (lane & 15);
  int coff = (lane >> 4) * 2;
  v2f ones = {1.0f, 1.0f};
  v8f acc = {};
  for (int c = cb; c < cb + 128; c += 4) {
    int cc = c + coff;
    const float2 vv = *(const float2*)(V + row * NMAT + cc);
    float g0 = good[cc];
    float g1 = good[cc + 1];
    float t0 = (vv.x != 0.0f) ? 1.0f - 0.5f * fabsf(vv.x - g0) : 0.0f;
    float t1 = (vv.y != 0.0f) ? 1.0f - 0.5f * fabsf(vv.y - g1) : 0.0f;
    v2f a = {t0, t1};
    acc = __builtin_amdgcn_wmma_f32_16x16x4_f32(false, a, false, ones,
                                                (short)0, acc, false, false);
  }
  if ((lane & 15) == 0) {
    int base = r0 + ((lane >> 4) << 3);
    atomicAdd(&Facc[base + 0], acc[0]);
    atomicAdd(&Facc[base + 1], acc[1]);
    atomicAdd(&Facc[base + 2], acc[2]);
    atomicAdd(&Facc[base + 3], acc[3]);
    atomicAdd(&Facc[base + 4], acc[4]);
    atomicAdd(&Facc[base + 5], acc[5]);
    atomicAdd(&Facc[base + 6], acc[6]);
    atomicAdd(&Facc[base + 7], acc[7]);
  }
}

// ------- fair[i] = clip(f[i] / (2048 + Noutp[i]), 0, 1)
__global__ void k_fair(const float* __restrict__ Facc, const float* __restrict__ Noutp,
                       float* out_fair) {
  int i = blockIdx.x * blockDim.x + threadIdx.x;
  if (i >= NMAT) return;
  float nout = 2048.0f + Noutp[i];
  float f = Facc[i] / nout;
  out_fair[i] = fminf(1.0f, fmaxf(0.0f, f));
}

// ------- gsum = sum over targets of good[t] (NaN -> 0), single block
__global__ void k_gsum(const float* __restrict__ good, const int* __restrict__ targets,
                       float* out0, int T) {
  __shared__ float sh[512];
  int t = threadIdx.x;
  float v = 0.0f;
  if (t < T) {
    float gv = good[targets[t]];
    v = (gv == gv) ? gv : 0.0f;
  }
  sh[t] = v;
  __syncthreads();
  for (int s = 256; s > 0; s >>= 1) {
    if (t < s) sh[t] += sh[t + s];
    __syncthreads();
  }
  if (t == 0) out0[0] = sh[0];
}

extern "C" void kernel_launch(void* const* d_in, const int* in_sizes, int n_in,
                              void* d_out, int out_size, void* d_ws, size_t ws_size,
                              hipStream_t stream) {
  const int* edges = (const int*)d_in[0];
  const float* weights = (const float*)d_in[1];
  const float* stat = (const float*)d_in[2];
  const int* edges_c = (const int*)d_in[3];
  const float* grdt = (const float*)d_in[4];
  const int* targets = (const int*)d_in[5];
  // d_in[6] = n (scalar, fixed 4096 here)

  int E = in_sizes[0] / 2;
  int EC = in_sizes[3] / 2;
  int T = in_sizes[5];

  float* A = (float*)d_ws;            // A counts -> then Sm05 (=S-0.5)
  float* W = A + MELEM;               // W sums
  float* Bv = W + MELEM;              // B sums -> then V (in place)
  float* Ninp = Bv + MELEM;           // 4096: colsum(S)-2048 partials
  float* Noutp = Ninp + NMAT;         // 4096: rowsum(S)-2048 partials
  float* Gacc = Noutp + NMAT;         // 4096: colsum(V)
  float* Facc = Gacc + NMAT;          // 4096: rowsum(contrib)

  size_t need = ((size_t)3 * MELEM + 4 * NMAT) * sizeof(float);
  if (ws_size < need) return;

  float* out = (float*)d_out;
  float* out_fair = out + 1;
  float* out_good = out + 1 + NMAT;

  int total_zero_f4 = (3 * MELEM + 4 * NMAT) / 4;
  k_zero_f4<<<4096, 256, 0, stream>>>((float4*)d_ws, total_zero_f4);
  k_scatter_edges<<<(E + 255) / 256, 256, 0, stream>>>(edges, weights, stat, A, W, E);
  k_scatter_edges_c<<<(EC + 255) / 256, 256, 0, stream>>>(edges_c, grdt, Bv, EC);
  k_combine<<<(MELEM / 4) / 256, 256, 0, stream>>>(A, W, Bv, MELEM / 4);
  // 8192 waves each: 256 col/row stripes x 32 blocks of 128
  k_colsum_wmma<<<1024, 256, 0, stream>>>(Bv, A, Gacc, Ninp);
  k_rowsum_wmma<<<1024, 256, 0, stream>>>(A, Noutp);
  k_good<<<NMAT / 256, 256, 0, stream>>>(Gacc, Ninp, out_good);
  k_fpass_wmma<<<1024, 256, 0, stream>>>(Bv, out_good, Facc);
  k_fair<<<NMAT / 256, 256, 0, stream>>>(Facc, Noutp, out_fair);
  k_gsum<<<1, 512, 0, stream>>>(out_good, targets, out, T);
}